// OuterProductMean_11570641895775
// MI455X (gfx1250) — compile-verified
//
#include <hip/hip_runtime.h>

typedef float v2f __attribute__((ext_vector_type(2)));
typedef float v8f __attribute__((ext_vector_type(8)));

#define DIM   384
#define RED   16
#define PAIR  128
#define MTOT  512
#define BTOT  2

// ---------------------------------------------------------------------------
// Kernel 1: fused LayerNorm + Linear(384 -> 16). One wave32 per (b,m) row.
// ---------------------------------------------------------------------------
__global__ void ln_reduce_kernel(const float* __restrict__ x,
                                 const float* __restrict__ gamma,
                                 const float* __restrict__ beta,
                                 const float* __restrict__ w_red,
                                 const float* __restrict__ b_red,
                                 float* __restrict__ xr) {
  const int row_id = (int)((blockIdx.x * blockDim.x + threadIdx.x) >> 5);
  const int lane   = (int)(threadIdx.x & 31);
  if (row_id >= BTOT * MTOT) return;

  const float* row = x + (size_t)row_id * DIM;

  // 384 / 32 = 12 elements per lane (strided by 32)
  float v[12];
  float s = 0.f, ss = 0.f;
#pragma unroll
  for (int t = 0; t < 12; ++t) {
    float f = row[lane + 32 * t];
    v[t] = f; s += f; ss += f * f;
  }
#pragma unroll
  for (int off = 16; off > 0; off >>= 1) {
    s  += __shfl_xor(s,  off, 32);
    ss += __shfl_xor(ss, off, 32);
  }
  const float mu   = s * (1.0f / DIM);
  const float var  = ss * (1.0f / DIM) - mu * mu;
  const float rstd = rsqrtf(var + 1e-5f);

  float acc[RED];
#pragma unroll
  for (int r = 0; r < RED; ++r) acc[r] = 0.f;
#pragma unroll
  for (int t = 0; t < 12; ++t) {
    const int j  = lane + 32 * t;
    const float xn = (v[t] - mu) * rstd * gamma[j] + beta[j];
#pragma unroll
    for (int r = 0; r < RED; ++r) acc[r] += xn * w_red[j * RED + r];
  }
  // cross-lane reduce each of the 16 outputs; lane r stores output r
#pragma unroll
  for (int r = 0; r < RED; ++r) {
    float a = acc[r];
#pragma unroll
    for (int off = 16; off > 0; off >>= 1) a += __shfl_xor(a, off, 32);
    if (lane == r) xr[row_id * RED + r] = a + b_red[r];
  }
}

// ---------------------------------------------------------------------------
// Kernel 2: U[b,n,i,p] = sum_j xr[b,n,j] * w_out[(i*16+j), p]   (8 MB, in ws)
// ---------------------------------------------------------------------------
__global__ void u_kernel(const float* __restrict__ xr,
                         const float* __restrict__ w_out,
                         float* __restrict__ U) {
  const int p  = (int)threadIdx.x;  // 0..127
  const int i  = (int)blockIdx.x;   // 0..15
  const int bn = (int)blockIdx.y;   // 0..1023
  float acc = 0.f;
#pragma unroll
  for (int j = 0; j < RED; ++j)
    acc += xr[bn * RED + j] * w_out[(i * RED + j) * PAIR + p];
  U[((size_t)bn * RED + i) * PAIR + p] = acc;
}

// ---------------------------------------------------------------------------
// Kernel 3: out[b,m,n,p] = sum_i xr[b,m,i] * U[b,n,i,p] + b_out[p]
// One block per (b,n); 8 waves, one per 16-wide p-tile; wave loops 32 m-tiles.
// Per tile: 4x V_WMMA_F32_16X16X4_F32 chained over K=16.
//
// ISA layouts (wave32, 7.12.2):
//   A 16x4 (MxK):  lane 0-15 -> M=lane,  VGPR0=K0,VGPR1=K1; lanes 16-31: K2,K3
//   B 4x16 (KxN):  VGPR0: K0 (lanes0-15,N=lane) / K2 (lanes16-31); VGPR1: K1/K3
//   C/D 16x16:     VGPR v: M=v (lanes0-15) / M=v+8 (lanes16-31); N=lane&15
// ---------------------------------------------------------------------------
__global__ void outer_wmma_kernel(const float* __restrict__ xr,
                                  const float* __restrict__ U,
                                  const float* __restrict__ b_out,
                                  float* __restrict__ out) {
  const int bn   = (int)blockIdx.x;        // 0..1023
  const int b    = bn >> 9;
  const int n    = bn & (MTOT - 1);
  const int wave = (int)(threadIdx.x >> 5); // 0..7  -> p tile
  const int lane = (int)(threadIdx.x & 31);
  const int col  = lane & 15;
  const int koff = (lane >> 4) * 2;        // 0 (lanes 0-15) or 2 (lanes 16-31)
  const int p_tile = wave * 16;

  // B fragments: reused across all 32 m-tiles
  v2f bf[4];
#pragma unroll
  for (int kk = 0; kk < 4; ++kk) {
    const float* ub = U + ((size_t)bn * RED + (kk * 4 + koff)) * PAIR + p_tile + col;
    bf[kk].x = ub[0];
    bf[kk].y = ub[PAIR];
  }
  const float bias = b_out[p_tile + col];

  const float* xrb  = xr + (size_t)b * MTOT * RED;
  float*       outb = out + (((size_t)b * MTOT) * MTOT + (size_t)n) * PAIR;

  for (int mt = 0; mt < MTOT / 16; ++mt) {
    // A fragments: row m = mt*16 + (lane&15), 2 consecutive K values per VGPR pair
    const float* ar = xrb + (size_t)(mt * 16 + col) * RED + koff;
    v2f af[4];
#pragma unroll
    for (int kk = 0; kk < 4; ++kk)
      af[kk] = *(const v2f*)(ar + kk * 4);

    // bias depends only on the column (p) -> same value in all 8 C VGPRs
    v8f c;
#pragma unroll
    for (int v = 0; v < 8; ++v) c[v] = bias;

#pragma unroll
    for (int kk = 0; kk < 4; ++kk) {
      c = __builtin_amdgcn_wmma_f32_16x16x4_f32(
          /*neg_a=*/false, af[kk], /*neg_b=*/false, bf[kk],
          /*c_mod=*/(short)0, c, /*reuse_a=*/false, /*reuse_b=*/false);
    }

    // store D: VGPR v -> row mt*16 + v (+8 for upper half-wave); streaming (NT)
    const int mrow = mt * 16 + ((lane >> 4) * 8);
#pragma unroll
    for (int v = 0; v < 8; ++v) {
      float* dst = outb + (size_t)(mrow + v) * (MTOT * PAIR) + p_tile + col;
      __builtin_nontemporal_store(c[v], dst);
    }
  }
}

// ---------------------------------------------------------------------------
extern "C" void kernel_launch(void* const* d_in, const int* in_sizes, int n_in,
                              void* d_out, int out_size, void* d_ws, size_t ws_size,
                              hipStream_t stream) {
  const float* x        = (const float*)d_in[0];
  const float* ln_gamma = (const float*)d_in[1];
  const float* ln_beta  = (const float*)d_in[2];
  const float* w_red    = (const float*)d_in[3];
  const float* b_red    = (const float*)d_in[4];
  const float* w_out    = (const float*)d_in[5];
  const float* b_out    = (const float*)d_in[6];
  float* out = (float*)d_out;

  float* xr = (float*)d_ws;                       // 1024*16 floats   (64 KB)
  float* U  = xr + (size_t)BTOT * MTOT * RED;     // 1024*16*128 floats (8 MB)

  // 1) LayerNorm + reduce: 1024 waves -> 128 blocks x 256 threads (8 waves)
  ln_reduce_kernel<<<dim3(128), dim3(256), 0, stream>>>(
      x, ln_gamma, ln_beta, w_red, b_red, xr);

  // 2) U = xr x w_out (per i): grid (i=16, bn=1024), 128 threads over p
  u_kernel<<<dim3(16, 1024), dim3(128), 0, stream>>>(xr, w_out, U);

  // 3) WMMA outer-product projection: one block per (b,n)
  outer_wmma_kernel<<<dim3(BTOT * MTOT), dim3(256), 0, stream>>>(
      xr, U, b_out, out);
}